// QSqueezeExcitation_7035156431137
// MI455X (gfx1250) — compile-verified
//
#include <hip/hip_runtime.h>

typedef __attribute__((ext_vector_type(2))) float v2f;
typedef __attribute__((ext_vector_type(8))) float v8f;

#define B_   32
#define C_   768
#define HW_  3136      // 56*56
#define RD_  192

// ---------------------------------------------------------------------------
// Kernel 1: global average pool.  One 256-thread block per (b,c) plane.
// ---------------------------------------------------------------------------
__global__ void __launch_bounds__(256) se_pool(const float* __restrict__ x,
                                               float* __restrict__ pooled) {
    const int bc = blockIdx.x;                       // 0 .. B_*C_-1
    const float4* p = (const float4*)(x + (size_t)bc * HW_);
    float sum = 0.0f;
    for (int i = threadIdx.x; i < HW_ / 4; i += 256) {
        float4 v = p[i];
        sum += v.x + v.y + v.z + v.w;
    }
    // wave32 reduction
    #pragma unroll
    for (int off = 16; off > 0; off >>= 1)
        sum += __shfl_down(sum, off, 32);
    __shared__ float wsum[8];
    const int lane = threadIdx.x & 31;
    const int wv   = threadIdx.x >> 5;
    if (lane == 0) wsum[wv] = sum;
    __syncthreads();
    if (threadIdx.x == 0) {
        float t = 0.0f;
        #pragma unroll
        for (int i = 0; i < 8; ++i) t += wsum[i];
        pooled[bc] = t * (1.0f / (float)HW_);
    }
}

// ---------------------------------------------------------------------------
// Kernel 2: the SE bottleneck, fully in f32 WMMA (V_WMMA_F32_16X16X4_F32).
//   h = relu(pooled [32,768] @ w1^T [768,192])      -> LDS
//   s = sigmoid(h   [32,192] @ w2^T [192,768])      -> global
// One workgroup, 768 threads = 24 wave32.
//
// ISA 7.12.2 layouts (f32, 16x16x4):
//   A (16x4):  lane l -> row m0+(l&15); VGPR0=K(k0+2*(l>>4)), VGPR1=K(+1)
//   B (4x16):  lane l -> col n0+(l&15); VGPR0=K(k0+2*(l>>4)), VGPR1=K(+1)
//   C/D:       VGPR v -> row m0+v+8*(l>>4), col n0+(l&15)
// ---------------------------------------------------------------------------
__global__ void __launch_bounds__(768) se_gemm(const float* __restrict__ pooled, // [32,768]
                                               const float* __restrict__ w1,     // [192,768]
                                               const float* __restrict__ w2,     // [768,192]
                                               float* __restrict__ s)            // [32,768]
{
    __shared__ __align__(16) float hbuf[B_ * RD_];   // 32*192*4 = 24 KB
    const int lane  = threadIdx.x & 31;
    const int wave  = threadIdx.x >> 5;              // 0..23
    const int l15   = lane & 15;
    const int khalf = (lane >> 4) * 2;               // 0 or 2
    const int radd  = (lane >> 4) * 8;               // C/D row offset

    // ---------------- GEMM1: 2x12 tiles, one per wave, K = 768 --------------
    {
        const int mt = wave / 12;                    // 0..1
        const int nt = wave % 12;                    // 0..11
        const int m0 = mt * 16;
        const int n0 = nt * 16;
        const float* arow = pooled + (size_t)(m0 + l15) * C_;   // A row
        const float* brow = w1     + (size_t)(n0 + l15) * C_;   // B^T row (w1[n,k])
        v8f acc = {};
        for (int k0 = 0; k0 < C_; k0 += 4) {
            const int kb = k0 + khalf;               // even -> 8B aligned
            v2f a = *(const v2f*)(arow + kb);
            v2f b = *(const v2f*)(brow + kb);
            acc = __builtin_amdgcn_wmma_f32_16x16x4_f32(
                false, a, false, b, (short)0, acc, false, false);
        }
        const int ncol = n0 + l15;
        #pragma unroll
        for (int v = 0; v < 8; ++v) {
            float val = acc[v];
            val = val > 0.0f ? val : 0.0f;           // ReLU
            hbuf[(m0 + v + radd) * RD_ + ncol] = val;
        }
    }
    __syncthreads();

    // ---------------- GEMM2: 2x48 tiles, 4 per wave, K = 192 ----------------
    for (int t = 0; t < 4; ++t) {
        const int tile = wave * 4 + t;               // 0..95
        const int mt = tile / 48;                    // 0..1
        const int nt = tile % 48;                    // 0..47
        const int m0 = mt * 16;
        const int n0 = nt * 16;
        const float* arow = hbuf + (m0 + l15) * RD_;            // h row (LDS)
        const float* brow = w2   + (size_t)(n0 + l15) * RD_;    // w2[n,k]
        v8f acc = {};
        for (int k0 = 0; k0 < RD_; k0 += 4) {
            const int kb = k0 + khalf;
            v2f a = *(const v2f*)(arow + kb);
            v2f b = *(const v2f*)(brow + kb);
            acc = __builtin_amdgcn_wmma_f32_16x16x4_f32(
                false, a, false, b, (short)0, acc, false, false);
        }
        const int ncol = n0 + l15;
        #pragma unroll
        for (int v = 0; v < 8; ++v) {
            float val = acc[v];
            val = 1.0f / (1.0f + __expf(-val));      // sigmoid
            s[(m0 + v + radd) * C_ + ncol] = val;
        }
    }
}

// ---------------------------------------------------------------------------
// Kernel 3: out = x * s (gate broadcast over H*W).  float4 per thread.
// ---------------------------------------------------------------------------
__global__ void __launch_bounds__(256) se_scale(const float* __restrict__ x,
                                                const float* __restrict__ s,
                                                float* __restrict__ out) {
    const size_t i4 = (size_t)blockIdx.x * 256 + threadIdx.x;   // float4 index
    const size_t bc = i4 / (HW_ / 4);                           // 784 float4 / plane
    const float sc = s[bc];
    float4 v = ((const float4*)x)[i4];
    v.x *= sc; v.y *= sc; v.z *= sc; v.w *= sc;
    ((float4*)out)[i4] = v;
}

// ---------------------------------------------------------------------------
extern "C" void kernel_launch(void* const* d_in, const int* in_sizes, int n_in,
                              void* d_out, int out_size, void* d_ws, size_t ws_size,
                              hipStream_t stream) {
    const float* x  = (const float*)d_in[0];   // [32,768,56,56]
    const float* w1 = (const float*)d_in[1];   // [192,768]
    const float* w2 = (const float*)d_in[2];   // [768,192]
    // d_in[3] = num_bits (0), d_in[4] = num_grad_bits (0): full-precision path.
    float* out    = (float*)d_out;
    float* pooled = (float*)d_ws;              // [32,768]
    float* sgate  = pooled + B_ * C_;          // [32,768]

    se_pool <<<B_ * C_, 256, 0, stream>>>(x, pooled);
    se_gemm <<<1, 768, 0, stream>>>(pooled, w1, w2, sgate);

    const int total4 = (B_ * C_ * HW_) / 4;    // 19,267,584
    se_scale<<<total4 / 256, 256, 0, stream>>>(x, sgate, out);
}